// Model_71270687310164
// MI455X (gfx1250) — compile-verified
//
#include <hip/hip_runtime.h>

typedef __attribute__((ext_vector_type(2))) float v2f;
typedef __attribute__((ext_vector_type(8))) float v8f;

#define N_NODES 50000
#define N_EDGES 800000

// ---------------- zero fill (graph-capture safe, no hipMemsetAsync) ----------
__global__ void zero_kernel(float4* __restrict__ p, int n4) {
    int tid = blockIdx.x * blockDim.x + threadIdx.x;
    if (tid < n4) p[tid] = make_float4(0.f, 0.f, 0.f, 0.f);
}

// ---------------- x = emb[n_id]  (float4 vectorized gather) -----------------
__global__ void gather_kernel(const int* __restrict__ n_id,
                              const float* __restrict__ emb,
                              float* __restrict__ x, int n) {
    int tid = blockIdx.x * blockDim.x + threadIdx.x;      // n*16 threads
    if (tid >= n * 16) return;
    int i = tid >> 4, g = (tid & 15) << 2;
    int s = n_id[i];
    *(float4*)(x + (size_t)i * 64 + g) =
        *(const float4*)(emb + (size_t)s * 64 + g);
}

// ---------------- in-degree via f32 atomics (exact: counts << 2^24) ---------
__global__ void deg_kernel(const int* __restrict__ dst,
                           float* __restrict__ deg, int nE) {
    int e = blockIdx.x * blockDim.x + threadIdx.x;
    if (e < nE) atomicAdd(&deg[dst[e]], 1.0f);
}

// ---------------- agg[dst,:] += xin[src,:]  (float4 read, 4 atomics) --------
template <int C>
__global__ void scatter_add_kernel(const float* __restrict__ xin,
                                   const int* __restrict__ src,
                                   const int* __restrict__ dst,
                                   float* __restrict__ agg, int nE) {
    constexpr int GSH = (C == 64) ? 4 : 5;                // log2(C/4)
    int gid = blockIdx.x * blockDim.x + threadIdx.x;      // nE * (C/4) threads
    int e = gid >> GSH;
    int g = (gid & ((C / 4) - 1)) << 2;
    if (e >= nE) return;
    int s = src[e], d = dst[e];
    float4 v = *(const float4*)(xin + (size_t)s * C + g);
    float* p = agg + (size_t)d * C + g;
    atomicAdd(p + 0, v.x);
    atomicAdd(p + 1, v.y);
    atomicAdd(p + 2, v.z);
    atomicAdd(p + 3, v.w);
}

// ------- out = act( (agg/deg) @ Wl + b + xin @ Wr ), fp32 WMMA 16x16x4 ------
// One wave per 16x16 output tile; K,N compile-time -> fully unrolled WMMA chain.
// M must be a multiple of 16 (50000 = 3125*16).
template <int K, int N, bool RELU>
__global__ __launch_bounds__(256) void sage_gemm_kernel(
        const float* __restrict__ agg, const float* __restrict__ deg,
        const float* __restrict__ xin, const float* __restrict__ Wl,
        const float* __restrict__ bias, const float* __restrict__ Wr,
        float* __restrict__ out, int M) {
    constexpr int TILES_N = N >> 4;
    const int lane   = threadIdx.x & 31;
    const int wave   = threadIdx.x >> 5;
    const int tile   = blockIdx.x * (blockDim.x >> 5) + wave;
    const int tile_m = tile / TILES_N;                 // constant divisor -> shift
    const int tile_n = tile - tile_m * TILES_N;
    if (tile_m * 16 >= M) return;                      // wave-uniform (EXEC all-1)

    const int half = lane >> 4;                        // 0: K+0/1, 1: K+2/3
    const int r    = lane & 15;
    const int rowA = tile_m * 16 + r;                  // row this lane feeds for A
    const int col  = tile_n * 16 + r;                  // col this lane feeds (B, D)

    const float rdeg = 1.0f / fmaxf(deg[rowA], 1.0f);
    const float* aggRow = agg + (size_t)rowA * K + 2 * half;
    const float* xRow   = xin + (size_t)rowA * K + 2 * half;
    const float* wlCol  = Wl + (size_t)(2 * half) * N + col;
    const float* wrCol  = Wr + (size_t)(2 * half) * N + col;

    v8f acc = {};
    #pragma unroll
    for (int k = 0; k < K; k += 4) {
        float2 ta = *(const float2*)(aggRow + k);
        float2 tx = *(const float2*)(xRow + k);
        v2f a1; a1.x = ta.x * rdeg; a1.y = ta.y * rdeg;
        v2f a2; a2.x = tx.x;        a2.y = tx.y;
        v2f b1; b1.x = wlCol[(size_t)k * N]; b1.y = wlCol[(size_t)(k + 1) * N];
        v2f b2; b2.x = wrCol[(size_t)k * N]; b2.y = wrCol[(size_t)(k + 1) * N];
        acc = __builtin_amdgcn_wmma_f32_16x16x4_f32(false, a1, false, b1,
                                                    (short)0, acc, false, false);
        acc = __builtin_amdgcn_wmma_f32_16x16x4_f32(false, a2, false, b2,
                                                    (short)0, acc, false, false);
    }

    const float bcol = bias[col];
    float* outBase = out + (size_t)(tile_m * 16 + 8 * half) * N + col;
    #pragma unroll
    for (int v = 0; v < 8; ++v) {                      // D: VGPR v -> row v + 8*half
        float val = acc[v] + bcol;
        if (RELU) val = fmaxf(val, 0.0f);
        outBase[(size_t)v * N] = val;
    }
}

// ---------------- out[e] = dot(h2[src], h2[dst]) over 64 ch, wave/edge ------
__global__ void edge_dot_kernel(const float* __restrict__ h2,
                                const int* __restrict__ src,
                                const int* __restrict__ dst,
                                float* __restrict__ out, int nE) {
    int gid  = blockIdx.x * blockDim.x + threadIdx.x;
    int e    = gid >> 5;
    int lane = gid & 31;
    if (e >= nE) return;
    int s = src[e], d = dst[e];
    float2 a = *(const float2*)(h2 + (size_t)s * 64 + lane * 2);
    float2 b = *(const float2*)(h2 + (size_t)d * 64 + lane * 2);
    float p = a.x * b.x + a.y * b.y;
    #pragma unroll
    for (int off = 16; off > 0; off >>= 1) p += __shfl_xor(p, off, 32);
    if (lane == 0) out[e] = p;
}

extern "C" void kernel_launch(void* const* d_in, const int* in_sizes, int n_in,
                              void* d_out, int out_size, void* d_ws, size_t ws_size,
                              hipStream_t stream) {
    const int*   n_id = (const int*)d_in[0];
    const int*   ei   = (const int*)d_in[1];         // [2, 800000]
    const float* emb  = (const float*)d_in[2];
    const float* W1l  = (const float*)d_in[3];
    const float* b1   = (const float*)d_in[4];
    const float* W1r  = (const float*)d_in[5];
    const float* W2l  = (const float*)d_in[6];
    const float* b2   = (const float*)d_in[7];
    const float* W2r  = (const float*)d_in[8];
    float*       out  = (float*)d_out;

    const int N = N_NODES, E = N_EDGES;
    const int* srcI = ei;
    const int* dstI = ei + E;

    // workspace layout (floats): deg | x | agg1 | h | h2 ; agg2 reuses x+agg1
    float* ws   = (float*)d_ws;
    float* deg  = ws;                         // 50000 (padded to 50048)
    float* x    = ws + 50048;                 // 50000*64 = 3,200,000
    float* agg1 = x + 3200000;                // 50000*64
    float* agg2 = x;                          // reuse [x|agg1] as 50000*128
    float* h    = agg1 + 3200000;             // 50000*128 = 6,400,000
    float* h2   = h + 6400000;                // 50000*64

    // zero deg + agg1 (contiguous: 50048 + 3,200,000 + 3,200,000 floats)
    {
        int n4 = (50048 + 3200000 + 3200000) / 4;
        zero_kernel<<<(n4 + 255) / 256, 256, 0, stream>>>((float4*)deg, n4);
    }
    gather_kernel<<<(N * 16) / 256, 256, 0, stream>>>(n_id, emb, x, N);
    deg_kernel<<<E / 256, 256, 0, stream>>>(dstI, deg, E);
    scatter_add_kernel<64><<<(E * 16) / 256, 256, 0, stream>>>(x, srcI, dstI,
                                                               agg1, E);
    // h = relu(agg1/deg @ W1l + b1 + x @ W1r)   [M=50000, K=64, N=128]
    sage_gemm_kernel<64, 128, true><<<3125, 256, 0, stream>>>(
        agg1, deg, x, W1l, b1, W1r, h, N);
    // zero agg2 (overwrites x|agg1, both dead now)
    {
        int n4 = 6400000 / 4;
        zero_kernel<<<(n4 + 255) / 256, 256, 0, stream>>>((float4*)agg2, n4);
    }
    scatter_add_kernel<128><<<(E * 32) / 256, 256, 0, stream>>>(h, srcI, dstI,
                                                                agg2, E);
    // h2 = agg2/deg @ W2l + b2 + h @ W2r        [M=50000, K=128, N=64]
    sage_gemm_kernel<128, 64, false><<<1563, 256, 0, stream>>>(
        agg2, deg, h, W2l, b2, W2r, h2, N);
    edge_dot_kernel<<<(E * 32) / 256, 256, 0, stream>>>(h2, srcI, dstI, out, E);
}